// CrossCovarianceAttention_46385646797523
// MI455X (gfx1250) — compile-verified
//
#include <hip/hip_runtime.h>
#include <math.h>
#include <stdint.h>

typedef __attribute__((ext_vector_type(2))) float v2f;
typedef __attribute__((ext_vector_type(8))) float v8f;
typedef __attribute__((ext_vector_type(4))) unsigned u32x4;
typedef __attribute__((ext_vector_type(8))) unsigned u32x8;

// D = A(16x4 f32) * B(4x16 f32) + C(16x16 f32)
#define WMMA_F32(a, b, c) \
  __builtin_amdgcn_wmma_f32_16x16x4_f32(false, (a), false, (b), (short)0, (c), false, false)

// ---------------------------------------------------------------------------
// TDM: DMA a 2D fp32 tile (nrows x rowlen) from global into LDS, with LDS row
// padding (pad_amt+1 DWORDs every 2^(pad_iv+1) DWORDs) for bank-conflict-free
// WMMA fragment reads. Tracked by TENSORcnt. 2D => descriptor groups 0+1 only.
// ---------------------------------------------------------------------------
__device__ __forceinline__ void tdm_load_2d(const float* gptr, unsigned lds_byte,
                                            unsigned rowlen, unsigned nrows,
                                            unsigned pad_iv, unsigned pad_amt,
                                            unsigned tensor_rows) {
  const unsigned long long ga = (unsigned long long)(uintptr_t)gptr;
  u32x4 g0;
  g0[0] = 1u;                                   // count=1, user mode, no gather
  g0[1] = lds_byte;                             // lds_addr [63:32]
  g0[2] = (unsigned)(ga & 0xFFFFFFFFu);         // global_addr [95:64]
  g0[3] = (unsigned)((ga >> 32) & 0x01FFFFFFu)  // global_addr [120:96]
        | (2u << 30);                           // type=2 (image)
  u32x8 g1;
  g1[0] = (2u << 16)                            // data_size = 4B
        | (1u << 20)                            // pad_enable
        | (pad_iv << 22) | (pad_amt << 25);     // LDS row padding
  g1[1] = (256u & 0xFFFFu) << 16;               // tensor_dim0 = 256 (row len)
  g1[2] = (tensor_rows & 0xFFFFu) << 16;        // tensor_dim1 lo
  g1[3] = (tensor_rows >> 16) | (rowlen << 16); // tensor_dim1 hi | tile_dim0
  g1[4] = nrows;                                // tile_dim1 (tile_dim2 = 0)
  g1[5] = 256u;                                 // tensor_dim0_stride lo
  g1[6] = 0u;
  g1[7] = 0u;
  asm volatile("tensor_load_to_lds %0, %1" :: "s"(g0), "s"(g1) : "memory");
}

// ---------------------------------------------------------------------------
// GEMM: C[8192,256] = A[8192,256] @ W[256,256] + bias[256]
// 256 threads = 8 waves; WG tile 64x64, each wave owns two 16x16 subtiles.
// A/W tiles staged by TDM (tensor_load_to_lds), double-buffered, K-chunk 32.
// ---------------------------------------------------------------------------
__global__ __launch_bounds__(256) void gemm256(const float* __restrict__ A,
                                               const float* __restrict__ W,
                                               const float* __restrict__ bias,
                                               float* __restrict__ C) {
  __shared__ float As[2][64][36];  // 64 rows x 32 k (+4 pad)
  __shared__ float Bs[2][32][68];  // 32 k x 64 n (+4 pad)
  const int tid  = threadIdx.x;
  const int wave = tid >> 5;
  const int lane = tid & 31;
  const int m0 = blockIdx.y * 64;
  const int n0 = blockIdx.x * 64;
  const int twRow = (wave & 3) * 16;
  const int twCol = (wave >> 2) * 32;
  const int lm = lane & 15;
  const int lk = (lane >> 4) << 1;

  v8f acc0 = {};
  v8f acc1 = {};

  if (wave == 0) {  // prefetch chunk 0 (TDM ignores EXEC; one issue per wave)
    tdm_load_2d(&A[(size_t)m0 * 256], (unsigned)(uintptr_t)&As[0][0][0],
                32u, 64u, 4u, 3u, 8192u);
    tdm_load_2d(&W[(size_t)n0], (unsigned)(uintptr_t)&Bs[0][0][0],
                64u, 32u, 5u, 3u, 256u);
  }

  for (int c = 0; c < 8; ++c) {
    const int cur = c & 1;
    if (wave == 0) {
      if (c < 7) {  // kick DMA for next chunk, then wait for current (in-order)
        const int k1 = (c + 1) * 32;
        tdm_load_2d(&A[(size_t)m0 * 256 + k1],
                    (unsigned)(uintptr_t)&As[cur ^ 1][0][0], 32u, 64u, 4u, 3u, 8192u);
        tdm_load_2d(&W[(size_t)k1 * 256 + n0],
                    (unsigned)(uintptr_t)&Bs[cur ^ 1][0][0], 64u, 32u, 5u, 3u, 256u);
        __builtin_amdgcn_s_wait_tensorcnt(2);
      } else {
        __builtin_amdgcn_s_wait_tensorcnt(0);
      }
    }
    __syncthreads();
#pragma unroll
    for (int kk = 0; kk < 32; kk += 4) {
      v2f a, b0, b1;
      a.x  = As[cur][twRow + lm][kk + lk];
      a.y  = As[cur][twRow + lm][kk + lk + 1];
      b0.x = Bs[cur][kk + lk][twCol + lm];
      b0.y = Bs[cur][kk + lk + 1][twCol + lm];
      b1.x = Bs[cur][kk + lk][twCol + 16 + lm];
      b1.y = Bs[cur][kk + lk + 1][twCol + 16 + lm];
      acc0 = WMMA_F32(a, b0, acc0);
      acc1 = WMMA_F32(a, b1, acc1);
    }
    __syncthreads();
  }

  const int mrow = m0 + twRow + ((lane >> 4) << 3);
#pragma unroll
  for (int r = 0; r < 8; ++r) {
    int gm  = mrow + r;
    int gn0 = n0 + twCol + lm;
    C[(size_t)gm * 256 + gn0]      = acc0[r] + bias[gn0];
    C[(size_t)gm * 256 + gn0 + 16] = acc1[r] + bias[gn0 + 16];
  }
}

// ---------------------------------------------------------------------------
// Attention: 128 (h2,b) pairs, each a contiguous [1024,16] slab of Q/K/V.
// One WG per (pair, 16-row q strip). Score strip [16x1024] kept in LDS.
// Exact lower median per row via 32-step binary search on ordered uint keys.
// ---------------------------------------------------------------------------
__global__ __launch_bounds__(256) void attn(const float* __restrict__ Qb,
                                            const float* __restrict__ Kb,
                                            const float* __restrict__ Vb,
                                            float* __restrict__ Ob) {
  __shared__ float S[16][1024];  // 64 KB score/prob strip
  const int tid  = threadIdx.x;
  const int wave = tid >> 5;
  const int lane = tid & 31;
  const int lm = lane & 15;
  const int lk = (lane >> 4) << 1;
  const int pair = blockIdx.x >> 6;  // 0..127 : h2*8 + b
  const int qblk = blockIdx.x & 63;  // 0..63  : 16-row strip
  const float* Q = Qb + (size_t)pair * 16384 + (size_t)qblk * 256;
  const float* K = Kb + (size_t)pair * 16384;
  const float* V = Vb + (size_t)pair * 16384;
  const float scale = 0.125f;  // 1/sqrt(DIM_K=64)

  __builtin_prefetch(V + (size_t)tid * 64, 0, 1);  // warm V slab (phase 3)

  // ---- Phase 1: scores = scale * Q Kt, strip [16 x 1024] -> LDS -----------
  v2f aq[4];
#pragma unroll
  for (int i = 0; i < 4; ++i) {
    aq[i].x = Q[lm * 16 + i * 4 + lk];
    aq[i].y = Q[lm * 16 + i * 4 + lk + 1];
  }
  for (int kt = wave; kt < 64; kt += 8) {
    v8f acc = {};
    const float* Kt = K + kt * 256;
#pragma unroll
    for (int i = 0; i < 4; ++i) {
      v2f b;
      b.x = Kt[lm * 16 + i * 4 + lk];
      b.y = Kt[lm * 16 + i * 4 + lk + 1];
      acc = WMMA_F32(aq[i], b, acc);
    }
    int col   = kt * 16 + lm;
    int rbase = (lane >> 4) << 3;
#pragma unroll
    for (int r = 0; r < 8; ++r) S[rbase + r][col] = acc[r] * scale;
  }
  __syncthreads();

  // ---- Phase 2: per-row lower median, mask, softmax (waves own 2 rows) ----
#pragma unroll
  for (int rr = 0; rr < 2; ++rr) {
    const int row = wave * 2 + rr;
    float vals[32];
    unsigned keys[32];
    float mx = -INFINITY;
#pragma unroll
    for (int i = 0; i < 32; ++i) {
      float v = S[row][lane + i * 32];
      vals[i] = v;
      unsigned u = __float_as_uint(v);
      keys[i] = (u & 0x80000000u) ? ~u : (u | 0x80000000u);  // order-preserving
      mx = fmaxf(mx, v);
    }
#pragma unroll
    for (int o = 16; o > 0; o >>= 1) mx = fmaxf(mx, __shfl_xor(mx, o, 32));

    unsigned lo = 0u, hi = 0xFFFFFFFFu;
    while (lo < hi) {  // 512-th smallest key == sorted[511] (lower median)
      unsigned mid = lo + ((hi - lo) >> 1);
      int cnt = 0;
#pragma unroll
      for (int i = 0; i < 32; ++i) cnt += (keys[i] <= mid) ? 1 : 0;
#pragma unroll
      for (int o = 16; o > 0; o >>= 1) cnt += __shfl_xor(cnt, o, 32);
      if (cnt >= 512) hi = mid; else lo = mid + 1;
    }
    const unsigned medkey = lo;

    float sum = 0.f;
#pragma unroll
    for (int i = 0; i < 32; ++i) {
      float e = (keys[i] > medkey) ? __expf(vals[i] - mx) : 0.f;
      vals[i] = e;
      sum += e;
    }
#pragma unroll
    for (int o = 16; o > 0; o >>= 1) sum += __shfl_xor(sum, o, 32);
    const float inv = 1.0f / sum;
#pragma unroll
    for (int i = 0; i < 32; ++i) S[row][lane + i * 32] = vals[i] * inv;
  }
  __syncthreads();

  // ---- Phase 3: O = P[16x1024] @ V[1024x16]; waves split K, reduce in LDS -
  v8f acc = {};
  const int kbeg = wave * 128;
  for (int k0 = kbeg; k0 < kbeg + 128; k0 += 4) {
    v2f a, b;
    a.x = S[lm][k0 + lk];
    a.y = S[lm][k0 + lk + 1];
    b.x = V[(k0 + lk) * 16 + lm];
    b.y = V[(k0 + lk + 1) * 16 + lm];
    acc = WMMA_F32(a, b, acc);
  }
  __syncthreads();                 // done reading S; reuse it for reduction
  float* Pacc = &S[0][0];          // [8][256]
#pragma unroll
  for (int r = 0; r < 8; ++r) Pacc[wave * 256 + r * 32 + lane] = acc[r];
  __syncthreads();

  {
    const int r = tid >> 5, l = tid & 31;
    float s = 0.f;
#pragma unroll
    for (int w = 0; w < 8; ++w) s += Pacc[w * 256 + r * 32 + l];
    const int Mrow = r + ((l >> 4) << 3);
    const int Ncol = l & 15;
    const int h2 = pair >> 3, b2 = pair & 7;
    const int qg = qblk * 16 + Mrow;
    // transpose(0,2,1,3) scatter: flat = h2*131072 + q*128 + b*16 + d
    Ob[(size_t)h2 * 131072 + (size_t)qg * 128 + b2 * 16 + Ncol] = s;
  }
}

// ---------------------------------------------------------------------------
extern "C" void kernel_launch(void* const* d_in, const int* in_sizes, int n_in,
                              void* d_out, int out_size, void* d_ws, size_t ws_size,
                              hipStream_t stream) {
  const float* x  = (const float*)d_in[0];
  const float* y  = (const float*)d_in[1];
  const float* Wq = (const float*)d_in[2];
  const float* bq = (const float*)d_in[3];
  const float* Wk = (const float*)d_in[4];
  const float* bk = (const float*)d_in[5];
  const float* Wv = (const float*)d_in[6];
  const float* bv = (const float*)d_in[7];
  const float* Wo = (const float*)d_in[8];
  const float* bo = (const float*)d_in[9];
  float* out = (float*)d_out;

  float* Qb = (float*)d_ws;          // 8 MB each
  float* Kb = Qb + 2097152;
  float* Vb = Kb + 2097152;
  float* Ob = Vb + 2097152;

  dim3 blk(256);
  dim3 gg(4, 128);                   // N tiles x M tiles (256/64, 8192/64)
  gemm256<<<gg, blk, 0, stream>>>(x, Wq, bq, Qb);
  gemm256<<<gg, blk, 0, stream>>>(y, Wk, bk, Kb);
  gemm256<<<gg, blk, 0, stream>>>(y, Wv, bv, Vb);
  attn<<<dim3(8192), blk, 0, stream>>>(Qb, Kb, Vb, Ob);
  gemm256<<<gg, blk, 0, stream>>>(Ob, Wo, bo, out);
}